// MoranIndexOptimized_20426864460202
// MI455X (gfx1250) — compile-verified
//
#include <hip/hip_runtime.h>
#include <math.h>

typedef float v2f __attribute__((ext_vector_type(2)));
typedef float v8f __attribute__((ext_vector_type(8)));

#define N_PTS 12288
#define KSEL 32
#define DECAY 0.1f
#define LOG2E 1.4426950408889634f
#define BLOCK_ROWS 64
#define THREADS 64
#define CW 128
#define SDPAD 4            // stride CW+4 keeps rows 16B-aligned for b128 scans
#define NCHUNK (N_PTS / CW)

// ws float accumulator layout:
// [0]=sum_y [1]=sum_yy [2]=sum_m [3]=sum_mm [4]=w_sum [5]=num_y [6]=num_m [7]=pad

__global__ void moran_init_acc(float* acc) {
  if (threadIdx.x < 8) acc[threadIdx.x] = 0.0f;
}

__global__ void moran_reduce_sums(const float* __restrict__ y,
                                  const float* __restrict__ m,
                                  float* acc) {
  int i = blockIdx.x * blockDim.x + threadIdx.x;
  float yv = 0.0f, mv = 0.0f;
  if (i < N_PTS) { yv = y[i]; mv = m[i]; }
  float sy = yv, syy = yv * yv, sm = mv, smm = mv * mv;
  // wave32 tree reduction
  for (int off = 16; off > 0; off >>= 1) {
    sy  += __shfl_down(sy,  off, 32);
    syy += __shfl_down(syy, off, 32);
    sm  += __shfl_down(sm,  off, 32);
    smm += __shfl_down(smm, off, 32);
  }
  if ((threadIdx.x & 31) == 0) {
    atomicAdd(&acc[0], sy);
    atomicAdd(&acc[1], syy);
    atomicAdd(&acc[2], sm);
    atomicAdd(&acc[3], smm);
  }
}

__global__ __launch_bounds__(THREADS)
void moran_main(const float4* __restrict__ tx, const float* __restrict__ y,
                const float* __restrict__ m, float* acc) {
  // d^2 tile; stride CW+4 -> rows 16B-aligned, bank = (4*row + col) % 64 on scans
  __shared__ float sdist[BLOCK_ROWS][CW + SDPAD];
  // padded top-k lists: stride 33 -> distinct banks across 32 lanes
  __shared__ float skd[BLOCK_ROWS][KSEL + 1];
  __shared__ int   ski[BLOCK_ROWS][KSEL + 1];

  const int t    = threadIdx.x;
  const int lane = t & 31;
  const int wv   = t >> 5;
  const int blockRow = blockIdx.x * BLOCK_ROWS;
  const int myRow    = blockRow + t;
  const int half = (lane >= 16) ? 1 : 0;
  const int l15  = lane & 15;

  // init own top-K list (only this thread reads/writes its row)
  for (int k = 0; k < KSEL; ++k) { skd[t][k] = 3.0e38f; ski[t][k] = 0; }

  // Load A fragments once: two 16-row tiles per wave, held in VGPRs all sweep.
  // A row layout per ISA: lanes 0-15 hold K=0,1 = {x,y}; lanes 16-31 hold K=2,3 = {sq,1}
  v2f aFrag[2];
#pragma unroll
  for (int a = 0; a < 2; ++a) {
    int p = blockRow + wv * 32 + a * 16 + l15;
    float4 P = tx[p];
    float sq = P.x * P.x + P.y * P.y;
    aFrag[a] = half ? (v2f){sq, 1.0f} : (v2f){P.x, P.y};
  }
  __syncthreads();

  for (int c = 0; c < NCHUNK; ++c) {
    const int C0 = c * CW;
#pragma unroll
    for (int bt = 0; bt < CW / 16; ++bt) {
      int q = C0 + bt * 16 + l15;
      float4 Q = tx[q];
      if (c + 1 < NCHUNK) __builtin_prefetch(&tx[q + CW], 0, 1);  // global_prefetch_b8
      float sqq = Q.x * Q.x + Q.y * Q.y;
      // B col layout per ISA: lanes 0-15 hold K=0,1 = {-2x,-2y}; lanes 16-31 hold K=2,3 = {1,sq}
      v2f bFrag = half ? (v2f){1.0f, sqq} : (v2f){-2.0f * Q.x, -2.0f * Q.y};
#pragma unroll
      for (int a = 0; a < 2; ++a) {
        v8f d = {};
        // D[i][j] = -2 x_i x_j - 2 y_i y_j + sq_i + sq_j  == squared distance tile
        d = __builtin_amdgcn_wmma_f32_16x16x4_f32(false, aFrag[a], false, bFrag,
                                                  (short)0, d, false, false);
        int rbase  = wv * 32 + a * 16 + half * 8;   // C/D layout: vgpr r -> row 8*half+r
        int clocal = bt * 16 + l15;
        // store raw d^2: selection is monotone under sqrt, defer sqrt to finalize
#pragma unroll
        for (int r = 0; r < 8; ++r) {
          sdist[rbase + r][clocal] = d[r];
        }
      }
    }
    __syncthreads();

    // Selection on d^2: thread t streams its own row, 4 candidates per LDS b128 load.
    float worst = skd[t][KSEL - 1];
    for (int cc = 0; cc < CW; cc += 4) {
      float4 dq = *(const float4*)&sdist[t][cc];
#pragma unroll
      for (int u = 0; u < 4; ++u) {
        float dv = (u == 0) ? dq.x : (u == 1) ? dq.y : (u == 2) ? dq.z : dq.w;
        if (dv < worst) {             // strict <: ties keep earlier index (matches top_k)
          int j = KSEL - 1;
          while (j > 0 && skd[t][j - 1] > dv) {
            skd[t][j] = skd[t][j - 1];
            ski[t][j] = ski[t][j - 1];
            --j;
          }
          skd[t][j] = dv;
          ski[t][j] = C0 + cc + u;
          worst = skd[t][KSEL - 1];
        }
      }
    }
    __syncthreads();
  }

  // Finalize: sqrt only the 32 survivors, then weights + Moran numerator terms.
  float sumY = acc[0], sumM = acc[2];
  float ybar = sumY / (float)N_PTS;
  float mbar = sumM / (float)N_PTS;
  float wloc = 0.0f, sy = 0.0f, sm = 0.0f;
#pragma unroll 4
  for (int k = 0; k < KSEL; ++k) {
    float dist = __builtin_amdgcn_sqrtf(fmaxf(skd[t][k], 1e-30f)); // v_sqrt_f32
    float wgt  = __builtin_amdgcn_exp2f(-DECAY * LOG2E * dist);    // v_exp_f32
    int j = ski[t][k];
    wloc += wgt;
    sy += wgt * (y[j] - ybar);
    sm += wgt * (m[j] - mbar);
  }
  float devY = y[myRow] - ybar;
  float devM = m[myRow] - mbar;
  atomicAdd(&acc[4], wloc);
  atomicAdd(&acc[5], devY * sy);
  atomicAdd(&acc[6], devM * sm);
}

__global__ void moran_finalize(const float* acc, float* out) {
  if (threadIdx.x == 0 && blockIdx.x == 0) {
    float denY = acc[1] - acc[0] * acc[0] / (float)N_PTS;  // sum yy - (sum y)^2/N
    float denM = acc[3] - acc[2] * acc[2] / (float)N_PTS;
    float scale = (float)N_PTS / acc[4];
    out[0] = scale * acc[5] / denY;
    out[1] = scale * acc[6] / denM;
  }
}

extern "C" void kernel_launch(void* const* d_in, const int* in_sizes, int n_in,
                              void* d_out, int out_size, void* d_ws, size_t ws_size,
                              hipStream_t stream) {
  const float4* tx = (const float4*)d_in[0];  // target_x (1,N,4) f32: use .x,.y
  const float*  y  = (const float*)d_in[1];   // target_y (1,N,1) f32
  const float*  m  = (const float*)d_in[2];   // mu       (1,N,1) f32
  float* acc = (float*)d_ws;
  float* out = (float*)d_out;

  moran_init_acc<<<1, 32, 0, stream>>>(acc);
  moran_reduce_sums<<<(N_PTS + 255) / 256, 256, 0, stream>>>(y, m, acc);
  moran_main<<<N_PTS / BLOCK_ROWS, THREADS, 0, stream>>>(tx, y, m, acc);
  moran_finalize<<<1, 32, 0, stream>>>(acc, out);
}